// KGAT_15994458210393
// MI455X (gfx1250) — compile-verified
//
#include <hip/hip_runtime.h>
#include <hip/hip_bf16.h>
#include <math.h>

// ---------------------------------------------------------------------------
// KGAT 2-layer GAT-style conv for MI455X (gfx1250, wave32).
//   N = 150015 nodes, D = 64, HEADS = 4, HD = 16, E = 2,000,000 edges.
// GEMM xw = x@W done with V_WMMA_F32_16X16X4_F32 (fp32-exact WMMA path).
// Edge phase is L2-resident (38MB feature table << 192MB L2).
// ---------------------------------------------------------------------------

#define KG_D      64
#define KG_HEADS  4
#define KG_HD     16
#define KG_NUSERS 100000
#define KG_NITEMS 50000
#define KG_NCATS  15
#define KG_N      (KG_NUSERS + KG_NITEMS + KG_NCATS)

typedef float v2f __attribute__((ext_vector_type(2)));
typedef float v8f __attribute__((ext_vector_type(8)));

// ---- float <-> ordered-uint key (for atomic float max via atomicMax(u32)) ----
__device__ __forceinline__ unsigned kg_fkey(float f) {
  unsigned u = __float_as_uint(f);
  return (u & 0x80000000u) ? ~u : (u | 0x80000000u);
}
__device__ __forceinline__ float kg_kdec(unsigned k) {
  unsigned u = (k & 0x80000000u) ? (k & 0x7FFFFFFFu) : ~k;
  return __uint_as_float(u);
}

// ---------------------------------------------------------------------------
// 0) build x0 = concat(user, item, cat); acc = x0
// ---------------------------------------------------------------------------
__global__ void kg_init(const float* __restrict__ ue, const float* __restrict__ ie,
                        const float* __restrict__ ce,
                        float* __restrict__ xcur, float* __restrict__ acc, long long nd) {
  long long t = (long long)blockIdx.x * blockDim.x + threadIdx.x;
  if (t >= nd) return;
  const long long uN = (long long)KG_NUSERS * KG_D;
  const long long iN = (long long)KG_NITEMS * KG_D;
  float v;
  if (t < uN)            v = ue[t];
  else if (t < uN + iN)  v = ie[t - uN];
  else                   v = ce[t - uN - iN];
  xcur[t] = v;
  acc[t]  = v;
}

// ---------------------------------------------------------------------------
// 1) xw = x @ W   (WMMA f32 16x16x4, fp32 exact)
//    One wave per 16x16 output tile. Block = 256 threads = 8 waves:
//    2 row-tiles x 4 col-tiles per block.
// A 16x4 f32 layout: lane<16 -> M=lane, {K=k,k+1}; lane>=16 -> M=lane-16, {K=k+2,k+3}
// B 4x16 f32 layout: mirrors A with N across lanes.
// C/D 16x16 f32: VGPR v, lanes 0-15 -> M=v,N=lane; lanes 16-31 -> M=v+8,N=lane-16
// ---------------------------------------------------------------------------
__global__ void kg_gemm_wmma(const float* __restrict__ x, const float* __restrict__ W,
                             float* __restrict__ xw, int n) {
  const int wave  = threadIdx.x >> 5;      // 0..7
  const int lane  = threadIdx.x & 31;
  const int cTile = wave & 3;              // 0..3  (D/16 = 4 col tiles)
  const int rTile = blockIdx.x * 2 + (wave >> 2);
  const int mBase = rTile * 16;
  if (mBase >= n) return;                  // uniform per wave
  const int nBase = cTile * 16;

  const int lh   = lane >> 4;              // 0/1 half-wave
  const int lm   = lane & 15;
  int arow = mBase + lm;
  if (arow >= n) arow = n - 1;             // clamp reads; tail stores masked below
  const float* __restrict__ xrow = x + (size_t)arow * KG_D;

  v8f c = {};
#pragma unroll
  for (int k = 0; k < KG_D; k += 4) {
    const int ka = k + lh * 2;
    v2f a, b;
    a.x = xrow[ka];
    a.y = xrow[ka + 1];
    b.x = W[(size_t)ka       * KG_D + nBase + lm];
    b.y = W[(size_t)(ka + 1) * KG_D + nBase + lm];
    c = __builtin_amdgcn_wmma_f32_16x16x4_f32(false, a, false, b, (short)0, c,
                                              false, false);
  }
#pragma unroll
  for (int v = 0; v < 8; ++v) {
    const int row = mBase + v + 8 * lh;
    if (row < n) xw[(size_t)row * KG_D + nBase + lm] = c[v];
  }
}

// ---------------------------------------------------------------------------
// 2) zero aggr + stats (aggr is nd floats, stats 8 dwords right after it)
// ---------------------------------------------------------------------------
__global__ void kg_zero(float* __restrict__ p, long long n) {
  long long t = (long long)blockIdx.x * blockDim.x + threadIdx.x;
  if (t < n) p[t] = 0.0f;
}

// ---------------------------------------------------------------------------
// 3) alpha[e,h] = leakyrelu( dot(xw[dst],a[h,:16]) + dot(xw[src],a[h,16:]) )
//    + wave-level max reduction per head -> atomicMax on ordered-uint keys
// ---------------------------------------------------------------------------
__global__ void kg_alpha(const float* __restrict__ xw, const float* __restrict__ a,
                         const int* __restrict__ src, const int* __restrict__ dst,
                         float* __restrict__ alpha, unsigned* __restrict__ maxkey,
                         long long EH) {
  long long t = (long long)blockIdx.x * blockDim.x + threadIdx.x;
  float val = -INFINITY;
  if (t < EH) {
    const int e = (int)(t >> 2);
    const int h = (int)(t & 3);
    const float* __restrict__ xd = xw + (size_t)dst[e] * KG_D + h * KG_HD;
    const float* __restrict__ xs = xw + (size_t)src[e] * KG_D + h * KG_HD;
    const float* __restrict__ ad = a + h * 2 * KG_HD;          // xi part
    const float* __restrict__ as = a + h * 2 * KG_HD + KG_HD;  // xj part
    float s = 0.0f;
#pragma unroll
    for (int i = 0; i < KG_HD; ++i) s += xd[i] * ad[i] + xs[i] * as[i];
    s = (s > 0.0f) ? s : 0.2f * s;  // LeakyReLU(0.2)
    alpha[t] = s;
    val = s;
  }
  // reduce max across lanes with the same (lane & 3) == head, wave32
#pragma unroll
  for (int m = 4; m < 32; m <<= 1) val = fmaxf(val, __shfl_xor(val, m, 32));
  const int lane = threadIdx.x & 31;
  if (lane < 4) atomicMax(&maxkey[lane], kg_fkey(val));
}

// ---------------------------------------------------------------------------
// 4) sum[h] += exp(alpha - max[h])  (wave reduce + float atomicAdd)
// ---------------------------------------------------------------------------
__global__ void kg_sumexp(const float* __restrict__ alpha,
                          const unsigned* __restrict__ maxkey,
                          float* __restrict__ sum, long long EH) {
  long long t = (long long)blockIdx.x * blockDim.x + threadIdx.x;
  float val = 0.0f;
  if (t < EH) {
    const int h = (int)(t & 3);
    val = __expf(alpha[t] - kg_kdec(maxkey[h]));
  }
#pragma unroll
  for (int m = 4; m < 32; m <<= 1) val += __shfl_xor(val, m, 32);
  const int lane = threadIdx.x & 31;
  if (lane < 4) atomicAdd(&sum[lane], val);
}

// ---------------------------------------------------------------------------
// 5) alpha <- exp(alpha - max) / sum   (softmax weights, stored in place)
// ---------------------------------------------------------------------------
__global__ void kg_norm(float* __restrict__ alpha, const unsigned* __restrict__ maxkey,
                        const float* __restrict__ sum, long long EH) {
  long long t = (long long)blockIdx.x * blockDim.x + threadIdx.x;
  if (t >= EH) return;
  const int h = (int)(t & 3);
  alpha[t] = __expf(alpha[t] - kg_kdec(maxkey[h])) / sum[h];
}

// ---------------------------------------------------------------------------
// 6) aggr[dst] += xw[src] * alpha   (float atomics; rows L2-resident)
//    thread t -> (edge = t>>6, channel = t&63); coalesced row reads.
// ---------------------------------------------------------------------------
__global__ void kg_scatter(const float* __restrict__ xw, const float* __restrict__ alpha,
                           const int* __restrict__ src, const int* __restrict__ dst,
                           float* __restrict__ aggr, long long total) {
  long long t = (long long)blockIdx.x * blockDim.x + threadIdx.x;
  if (t >= total) return;
  const int e = (int)(t >> 6);
  const int d = (int)(t & 63);
  const int h = d >> 4;
  const float w = alpha[(size_t)e * 4 + h];
  const float v = xw[(size_t)src[e] * KG_D + d] * w;
  atomicAdd(&aggr[(size_t)dst[e] * KG_D + d], v);
}

// ---------------------------------------------------------------------------
// 7) xcur = elu(aggr); acc += xcur
// ---------------------------------------------------------------------------
__global__ void kg_elu_acc(const float* __restrict__ aggr, float* __restrict__ xcur,
                           float* __restrict__ acc, long long nd) {
  long long t = (long long)blockIdx.x * blockDim.x + threadIdx.x;
  if (t >= nd) return;
  const float v = aggr[t];
  const float y = (v > 0.0f) ? v : (__expf(v) - 1.0f);
  xcur[t] = y;
  acc[t] += y;
}

// ---------------------------------------------------------------------------
// 8) out = acc[:NUSERS+NITEMS] / 3
// ---------------------------------------------------------------------------
__global__ void kg_final(const float* __restrict__ acc, float* __restrict__ out,
                         long long n) {
  long long t = (long long)blockIdx.x * blockDim.x + threadIdx.x;
  if (t < n) out[t] = acc[t] * (1.0f / 3.0f);
}

// ---------------------------------------------------------------------------
extern "C" void kernel_launch(void* const* d_in, const int* in_sizes, int n_in,
                              void* d_out, int out_size, void* d_ws, size_t ws_size,
                              hipStream_t stream) {
  const float* ue = (const float*)d_in[0];
  const float* ie = (const float*)d_in[1];
  const float* ce = (const float*)d_in[2];
  const float* W1 = (const float*)d_in[3];
  const float* a1 = (const float*)d_in[4];
  const float* W2 = (const float*)d_in[5];
  const float* a2 = (const float*)d_in[6];
  const int*   ei = (const int*)d_in[7];

  const long long E   = in_sizes[7] / 2;       // 2,000,000
  const int*      src = ei;
  const int*      dst = ei + E;

  const long long nd  = (long long)KG_N * KG_D;          // node features
  const long long EH  = E * KG_HEADS;                    // edge-head count
  const long long ED  = E * KG_D;                        // edge-channel count
  const long long onf = (long long)(KG_NUSERS + KG_NITEMS) * KG_D;

  // ---- workspace carve-up (floats) ----
  float*    ws    = (float*)d_ws;
  float*    xcur  = ws;                 // [N*D]
  float*    acc   = xcur + nd;          // [N*D]
  float*    xw    = acc + nd;           // [N*D]
  float*    aggr  = xw + nd;            // [N*D]
  unsigned* stats = (unsigned*)(aggr + nd);   // [8]: 0..3 maxkey, 4..7 sum
  float*    sum   = (float*)(stats + 4);
  float*    alpha = (float*)(stats + 8);      // [E*4]

  const int B = 256;
  const int rowTiles  = (KG_N + 15) / 16;
  const int gemmBlks  = (rowTiles + 1) / 2;   // 2 row-tiles per block
  auto blks = [&](long long n) { return (unsigned)((n + B - 1) / B); };

  kg_init<<<blks(nd), B, 0, stream>>>(ue, ie, ce, xcur, acc, nd);

  const float* Ws[2] = {W1, W2};
  const float* As[2] = {a1, a2};
  for (int layer = 0; layer < 2; ++layer) {
    kg_gemm_wmma<<<gemmBlks, B, 0, stream>>>(xcur, Ws[layer], xw, KG_N);
    // zero aggr + the 8 stat dwords that sit immediately after it
    kg_zero<<<blks(nd + 8), B, 0, stream>>>(aggr, nd + 8);
    kg_alpha<<<blks(EH), B, 0, stream>>>(xw, As[layer], src, dst, alpha, stats, EH);
    kg_sumexp<<<blks(EH), B, 0, stream>>>(alpha, stats, sum, EH);
    kg_norm<<<blks(EH), B, 0, stream>>>(alpha, stats, sum, EH);
    kg_scatter<<<blks(ED), B, 0, stream>>>(xw, alpha, src, dst, aggr, ED);
    kg_elu_acc<<<blks(nd), B, 0, stream>>>(aggr, xcur, acc, nd);
  }

  kg_final<<<blks(onf), B, 0, stream>>>(acc, (float*)d_out, onf);
}